// G_GIN_45561013076148
// MI455X (gfx1250) — compile-verified
//
#include <hip/hip_runtime.h>
#include <hip/hip_bf16.h>

#define N_NODES 50000
#define N_EDGES 800000
#define N_GRAPHS 500
#define DIM 128
#define BN_EPS_F 1e-5f

typedef __attribute__((ext_vector_type(2))) float v2f;
typedef __attribute__((ext_vector_type(8))) float v8f;

// ---------------------------------------------------------------- utilities
__global__ void zero_f(float* __restrict__ p, int n) {
  int i = blockIdx.x * blockDim.x + threadIdx.x;
  if (i < n) p[i] = 0.0f;
}

// ------------------------------------------------------- edge scatter (aggr)
// One wave (32 lanes) per edge; each lane handles 4 features via float4.
// aggr[dst] += ew[e] * h[src]   (coalesced gather, atomic scatter)
__global__ __launch_bounds__(256) void scatter_edges(
    const int* __restrict__ ei, const float* __restrict__ h,
    const float* __restrict__ ew, float* __restrict__ aggr) {
  int wave = threadIdx.x >> 5;
  int lane = threadIdx.x & 31;
  int e = blockIdx.x * 8 + wave;           // E == 100000 * 8 exactly
  int src = ei[e];
  int dst = ei[N_EDGES + e];
  float w = ew[e];
  const float4* xs = (const float4*)(h + (size_t)src * DIM);
  float4 v = xs[lane];
  float* ad = aggr + (size_t)dst * DIM + lane * 4;
  atomicAdd(ad + 0, v.x * w);
  atomicAdd(ad + 1, v.y * w);
  atomicAdd(ad + 2, v.z * w);
  atomicAdd(ad + 3, v.w * w);
}

// --------------------------------------------- GIN linear via fp32 WMMA
// out[16-row tile] = ((1+eps)*h + aggr) @ W^T + b
// Block: 256 threads = 8 waves; wave y owns output columns [16y,16y+16).
// A-frag (16x4 f32, 2 VGPR): lane&15 = M, K pair = k0 + 2*(lane>>4)
// B-frag (4x16 f32, 2 VGPR): lane&15 = N, K pair = k0 + 2*(lane>>4)
// C/D (16x16 f32, 8 VGPR):   M = v + 8*(lane>>4), N = lane&15
__global__ __launch_bounds__(256) void gin_gemm(
    const float* __restrict__ h, const float* __restrict__ aggr,
    const float* __restrict__ eps_ptr, int eps_idx,
    const float* __restrict__ W, const float* __restrict__ bias,
    float* __restrict__ out) {
  __shared__ float Zs[16][DIM + 4];        // +4 pad: K-strided reads conflict-free
  int tid = threadIdx.y * 32 + threadIdx.x;
  int row0 = blockIdx.x * 16;              // N_NODES == 3125 * 16 exactly
  float epsv = 1.0f + eps_ptr[eps_idx];

  for (int i = tid; i < 16 * DIM; i += 256) {
    int r = i >> 7, c = i & 127;
    size_t g = (size_t)(row0 + r) * DIM + c;
    Zs[r][c] = epsv * h[g] + aggr[g];
  }
  __syncthreads();

  int lane = threadIdx.x;
  int col0 = threadIdx.y * 16;
  int m = lane & 15;
  int khalf = (lane >> 4) * 2;
  const float* wr_base = W + (size_t)(col0 + m) * DIM;  // B[k][n] = W[n][k]

  v8f acc = {};
  #pragma unroll 4
  for (int k0 = 0; k0 < DIM; k0 += 4) {
    v2f a, b;
    a.x = Zs[m][k0 + khalf];
    a.y = Zs[m][k0 + khalf + 1];
    b.x = wr_base[k0 + khalf];
    b.y = wr_base[k0 + khalf + 1];
    acc = __builtin_amdgcn_wmma_f32_16x16x4_f32(
        false, a, false, b, (short)0, acc, false, false);
  }

  int n = col0 + m;
  float bn = bias[n];
  int mbase = (lane >> 4) * 8;
  #pragma unroll
  for (int v = 0; v < 8; ++v) {
    out[(size_t)(row0 + mbase + v) * DIM + n] = acc[v] + bn;
  }
}

// --------------------------------------------------------- batchnorm stats
// Per-feature sum and sum-of-squares over all N nodes -> stats[0:128]=sum,
// stats[128:256]=sumsq. Each block handles a chunk of rows; thread = feature.
#define STAT_ROWS 128
__global__ __launch_bounds__(128) void bn_stats(
    const float* __restrict__ h, float* __restrict__ stats) {
  int f = threadIdx.x;
  int r0 = blockIdx.x * STAT_ROWS;
  int r1 = r0 + STAT_ROWS;
  if (r1 > N_NODES) r1 = N_NODES;
  float s = 0.0f, sq = 0.0f;
  for (int r = r0; r < r1; ++r) {
    float v = h[(size_t)r * DIM + f];
    s += v;
    sq += v * v;
  }
  atomicAdd(&stats[f], s);
  atomicAdd(&stats[DIM + f], sq);
}

// ----------------------------------------------------- batchnorm + relu
__global__ void bn_relu(float* __restrict__ h, const float* __restrict__ stats,
                        const float* __restrict__ gamma,
                        const float* __restrict__ beta) {
  int i = blockIdx.x * blockDim.x + threadIdx.x;
  if (i >= N_NODES * DIM) return;
  int f = i & 127;
  const float invn = 1.0f / (float)N_NODES;
  float mu = stats[f] * invn;
  float var = stats[DIM + f] * invn - mu * mu;   // population var, matches jnp.var
  float y = (h[i] - mu) * rsqrtf(var + BN_EPS_F) * gamma[f] + beta[f];
  h[i] = fmaxf(y, 0.0f);
}

// -------------------------------------------------------- mean pool scatter
__global__ __launch_bounds__(128) void pool_scatter(
    const float* __restrict__ h, const int* __restrict__ batch,
    float* __restrict__ sums, float* __restrict__ cnt) {
  int node = blockIdx.x;
  int f = threadIdx.x;
  int g = batch[node];
  atomicAdd(&sums[(size_t)g * DIM + f], h[(size_t)node * DIM + f]);
  if (f == 0) atomicAdd(&cnt[g], 1.0f);
}

// ----------------------------------------------------- final projection
// out[g][o] = bp[o] + sum_k (sums[g][k]/max(cnt[g],1)) * Wp[o][k]
__global__ void final_proj(const float* __restrict__ sums,
                           const float* __restrict__ cnt,
                           const float* __restrict__ Wp,
                           const float* __restrict__ bp,
                           float* __restrict__ out) {
  int i = blockIdx.x * blockDim.x + threadIdx.x;  // g*128 + o
  if (i >= N_GRAPHS * DIM) return;
  int g = i >> 7, o = i & 127;
  float inv = 1.0f / fmaxf(cnt[g], 1.0f);
  const float* sg = sums + (size_t)g * DIM;
  const float* wr = Wp + (size_t)o * DIM;
  float acc = bp[o];
  #pragma unroll 8
  for (int k = 0; k < DIM; ++k) acc += sg[k] * inv * wr[k];
  out[i] = acc;
}

// ---------------------------------------------------------------- driver
extern "C" void kernel_launch(void* const* d_in, const int* in_sizes, int n_in,
                              void* d_out, int out_size, void* d_ws, size_t ws_size,
                              hipStream_t stream) {
  const int*   edge_index = (const int*)d_in[0];     // [2,E]
  const float* x          = (const float*)d_in[1];   // [N,128]
  const int*   batch      = (const int*)d_in[2];     // [N]
  const float* ew         = (const float*)d_in[3];   // [E]
  const float* W0 = (const float*)d_in[4];
  const float* b0 = (const float*)d_in[5];
  const float* W1 = (const float*)d_in[6];
  const float* b1 = (const float*)d_in[7];
  const float* W2 = (const float*)d_in[8];
  const float* b2 = (const float*)d_in[9];
  const float* eps      = (const float*)d_in[10];    // [3]
  const float* bn_gamma = (const float*)d_in[11];    // [2,128]
  const float* bn_beta  = (const float*)d_in[12];    // [2,128]
  const float* Wp = (const float*)d_in[13];
  const float* bp = (const float*)d_in[14];
  float* out = (float*)d_out;                         // [500,128]

  const size_t NB = (size_t)N_NODES * DIM * sizeof(float);  // 25.6 MB
  char* ws = (char*)d_ws;
  float* aggr  = (float*)(ws);                     // [N,128]
  float* hb    = (float*)(ws + NB);                // [N,128]
  float* ha    = (float*)(ws + 2 * NB);            // [N,128]
  float* stats = (float*)(ws + 3 * NB);            // [256]
  float* sums  = (float*)(ws + 3 * NB + 1024);     // [G,128]
  float* cnt   = (float*)(ws + 3 * NB + 1024 + (size_t)N_GRAPHS * DIM * sizeof(float));

  const int NF = N_NODES * DIM;                    // 6,400,000
  dim3 gemm_block(32, 8);
  int gemm_grid = N_NODES / 16;                    // 3125
  int scat_grid = N_EDGES / 8;                     // 100000
  int stat_grid = (N_NODES + STAT_ROWS - 1) / STAT_ROWS;

  // ---- layer 0: conv -> BN -> relu
  zero_f<<<(NF + 255) / 256, 256, 0, stream>>>(aggr, NF);
  scatter_edges<<<scat_grid, 256, 0, stream>>>(edge_index, x, ew, aggr);
  gin_gemm<<<gemm_grid, gemm_block, 0, stream>>>(x, aggr, eps, 0, W0, b0, hb);
  zero_f<<<1, 256, 0, stream>>>(stats, 256);
  bn_stats<<<stat_grid, 128, 0, stream>>>(hb, stats);
  bn_relu<<<(NF + 255) / 256, 256, 0, stream>>>(hb, stats, bn_gamma, bn_beta);

  // ---- layer 1: conv -> BN -> relu
  zero_f<<<(NF + 255) / 256, 256, 0, stream>>>(aggr, NF);
  scatter_edges<<<scat_grid, 256, 0, stream>>>(edge_index, hb, ew, aggr);
  gin_gemm<<<gemm_grid, gemm_block, 0, stream>>>(hb, aggr, eps, 1, W1, b1, ha);
  zero_f<<<1, 256, 0, stream>>>(stats, 256);
  bn_stats<<<stat_grid, 128, 0, stream>>>(ha, stats);
  bn_relu<<<(NF + 255) / 256, 256, 0, stream>>>(ha, stats, bn_gamma + DIM, bn_beta + DIM);

  // ---- layer 2: conv only
  zero_f<<<(NF + 255) / 256, 256, 0, stream>>>(aggr, NF);
  scatter_edges<<<scat_grid, 256, 0, stream>>>(edge_index, ha, ew, aggr);
  gin_gemm<<<gemm_grid, gemm_block, 0, stream>>>(ha, aggr, eps, 2, W2, b2, hb);

  // ---- global mean pool + projection
  int pool_elems = N_GRAPHS * DIM + N_GRAPHS;      // sums then cnt (contiguous)
  zero_f<<<(pool_elems + 255) / 256, 256, 0, stream>>>(sums, pool_elems);
  pool_scatter<<<N_NODES, 128, 0, stream>>>(hb, batch, sums, cnt);
  final_proj<<<(N_GRAPHS * DIM + 255) / 256, 256, 0, stream>>>(sums, cnt, Wp, bp, out);
}